// Brain_73942156967974
// MI455X (gfx1250) — compile-verified
//
#include <hip/hip_runtime.h>
#include <stdint.h>

#define STEPS   3
#define WAVE    32
#define WPB     8                 // waves per block
#define BLOCK   (WAVE * WPB)      // 256 threads
#define TEDGES  128               // edges per wave-tile (32 lanes x 4 edges)
#define DEPTH   6                 // LDS slots per wave (pipeline depth)
#define AHEAD   (DEPTH - 1)       // tiles issued ahead of consumption

// -------- gfx1250 async global->LDS copy (ASYNCcnt path) --------------------
#define HAS_ASYNC (__has_builtin(__builtin_amdgcn_global_load_async_to_lds_b128))

typedef int v4i __attribute__((ext_vector_type(4)));
typedef __attribute__((address_space(1))) v4i* gas_v4i;
typedef __attribute__((address_space(3))) v4i* las_v4i;

__device__ __forceinline__ void async_cp16(const void* g, void* l) {
#if HAS_ASYNC
  // cpol = 1 -> TH=NT: keep the 240MB/step edge streams from thrashing L2,
  // so the 4MB v/acc state stays L2-resident for gathers/atomics.
  __builtin_amdgcn_global_load_async_to_lds_b128(
      (gas_v4i)g, (las_v4i)l, 0, 1);
#else
  *(int4*)l = *(const int4*)g;    // synchronous fallback
#endif
}

#if HAS_ASYNC
# if __has_builtin(__builtin_amdgcn_s_wait_asynccnt)
#  define WAIT_ASYNC(n) __builtin_amdgcn_s_wait_asynccnt(n)
# else
#  define WAIT_ASYNC(n) asm volatile("s_wait_asynccnt %0" :: "n"(n) : "memory")
# endif
#else
# define WAIT_ASYNC(n) do {} while (0)
#endif

__device__ __forceinline__ void atomic_add_f32(float* p, float v) {
  __hip_atomic_fetch_add(p, v, __ATOMIC_RELAXED, __HIP_MEMORY_SCOPE_AGENT);
}

// -------- main edge kernel: async-pipelined gather * w -> scatter-add -------
// All edge-stream offsets fit in 32 bits (E*4B = 80MB), so the hot loop uses
// uint byte offsets off SGPR base pointers (GVS saddr+voffset addressing).
__global__ void __launch_bounds__(BLOCK)
brain_edges(const float* __restrict__ v,
            const float* __restrict__ wgt,
            const int*   __restrict__ src,
            const int*   __restrict__ dst,
            float*       __restrict__ acc,
            int nTiles)
{
  // per-wave private slots: DEPTH x {src[128], dst[128], w[128]}
  __shared__ __align__(16) int lds[WPB * DEPTH * 3 * TEDGES];
  const int lane = threadIdx.x & (WAVE - 1);
  const int wid  = threadIdx.x / WAVE;
  int* wbuf = lds + wid * (DEPTH * 3 * TEDGES);

  const int gw = blockIdx.x * WPB + wid;      // global wave id
  const int nw = gridDim.x * WPB;             // total waves
  const int iters = (nTiles + nw - 1) / nw;   // uniform per wave

  const unsigned laneB = (unsigned)lane * 16u;
  const char* srcB = (const char*)src;
  const char* dstB = (const char*)dst;
  const char* wgtB = (const char*)wgt;

  auto issue = [&](int j) {
    int t = gw + j * nw;
    // dummy-but-valid address keeps ASYNCcnt bookkeeping uniform at the tail
    unsigned off = (t < nTiles) ? (unsigned)t * (TEDGES * 4u) + laneB : laneB;
    int* slot = wbuf + (j % DEPTH) * (3 * TEDGES);
    async_cp16(srcB + off, (char*)(slot)              + laneB);
    async_cp16(dstB + off, (char*)(slot + TEDGES)     + laneB);
    async_cp16(wgtB + off, (char*)(slot + 2 * TEDGES) + laneB);
  };

  for (int k = 0; k < AHEAD; ++k) issue(k);   // prologue

  for (int j = 0; j < iters; ++j) {
    issue(j + AHEAD);              // refill slot consumed last iteration
    WAIT_ASYNC(3 * AHEAD);         // oldest tile (j) fully landed in LDS
    int t = gw + j * nw;
    if (t < nTiles) {
      int* slot = wbuf + (j % DEPTH) * (3 * TEDGES);
      int4   s4 = ((const int4*)  (slot))[lane];
      int4   d4 = ((const int4*)  (slot + TEDGES))[lane];
      float4 w4 = ((const float4*)(slot + 2 * TEDGES))[lane];
      // gathers hit L2-resident v; atomics hit L2-resident acc
      atomic_add_f32(acc + d4.x, v[s4.x] * w4.x);
      atomic_add_f32(acc + d4.y, v[s4.y] * w4.y);
      atomic_add_f32(acc + d4.z, v[s4.z] * w4.z);
      atomic_add_f32(acc + d4.w, v[s4.w] * w4.w);
    }
  }
  WAIT_ASYNC(0);
}

// -------- tail: E % 128 edges with direct (non-temporal) loads --------------
__global__ void brain_edges_tail(const float* __restrict__ v,
                                 const float* __restrict__ wgt,
                                 const int*   __restrict__ src,
                                 const int*   __restrict__ dst,
                                 float*       __restrict__ acc,
                                 long long start, long long E)
{
  long long i = start + (long long)blockIdx.x * blockDim.x + threadIdx.x;
  if (i < E) {
    int   s = __builtin_nontemporal_load(src + i);
    int   d = __builtin_nontemporal_load(dst + i);
    float w = __builtin_nontemporal_load(wgt + i);
    atomic_add_f32(acc + d, v[s] * w);
  }
}

// -------- small service kernels --------------------------------------------
__global__ void init_state(float* v, float* acc, unsigned char* m, int n) {
  int i = blockIdx.x * blockDim.x + threadIdx.x;
  if (i < n) { v[i] = 0.f; acc[i] = 0.f; m[i] = 0; }
}
__global__ void set_inputs(const float* __restrict__ x, const int* __restrict__ idx,
                           float* v, int n) {
  int i = blockIdx.x * blockDim.x + threadIdx.x;
  if (i < n) v[idx[i]] = x[i];
}
__global__ void set_mask(const int* __restrict__ idx, unsigned char* m, int n) {
  int i = blockIdx.x * blockDim.x + threadIdx.x;
  if (i < n) m[idx[i]] = 1;
}
__global__ void add_bias(const int* __restrict__ nii, const float* __restrict__ b,
                         float* __restrict__ acc, int n) {
  int i = blockIdx.x * blockDim.x + threadIdx.x;
  if (i < n) acc[nii[i]] += b[i];          // unique indices: plain RMW is safe
}
__global__ void update_state(float* __restrict__ v, float* __restrict__ acc,
                             const unsigned char* __restrict__ omask, int n) {
  int i = blockIdx.x * blockDim.x + threadIdx.x;
  if (i < n) {
    float val = acc[i];
    acc[i] = 0.f;                          // re-arm accumulator for next step
    v[i] = omask[i] ? val : tanhf(val);
  }
}
__global__ void gather_out(const float* __restrict__ v, const int* __restrict__ idx,
                           float* __restrict__ out, int n) {
  int i = blockIdx.x * blockDim.x + threadIdx.x;
  if (i < n) out[i] = v[idx[i]];
}

// ---------------------------------------------------------------------------
extern "C" void kernel_launch(void* const* d_in, const int* in_sizes, int n_in,
                              void* d_out, int out_size, void* d_ws, size_t ws_size,
                              hipStream_t stream)
{
  const float* x    = (const float*)d_in[0];
  const float* wgt  = (const float*)d_in[1];
  const float* bias = (const float*)d_in[2];
  const int*   src  = (const int*)d_in[3];
  const int*   dst  = (const int*)d_in[4];
  const int*   iin  = (const int*)d_in[5];
  const int*   iout = (const int*)d_in[6];
  const int*   inon = (const int*)d_in[7];
  float* out = (float*)d_out;

  const long long IN  = in_sizes[0];
  const long long E   = in_sizes[1];
  const long long NB  = in_sizes[2];          // N - IN (biases)
  const long long N   = NB + IN;
  const long long NO  = in_sizes[6];          // OUT

  float* v            = (float*)d_ws;
  float* acc          = v + N;
  unsigned char* msk  = (unsigned char*)(acc + N);

  const int tpb = 256;
  init_state<<<(int)((N + tpb - 1) / tpb), tpb, 0, stream>>>(v, acc, msk, (int)N);
  set_inputs<<<(int)((IN + tpb - 1) / tpb), tpb, 0, stream>>>(x, iin, v, (int)IN);
  set_mask  <<<(int)((NO + tpb - 1) / tpb), tpb, 0, stream>>>(iout, msk, (int)NO);

  const long long nTiles = E / TEDGES;
  const long long tail0  = nTiles * TEDGES;
  const long long rem    = E - tail0;

  int egrid = 1024;                            // 8192 waves in flight
  if (nTiles < (long long)egrid * WPB)
    egrid = (int)((nTiles + WPB - 1) / WPB);
  if (egrid < 1) egrid = 1;

  for (int s = 0; s < STEPS; ++s) {
    if (nTiles > 0)
      brain_edges<<<egrid, BLOCK, 0, stream>>>(v, wgt, src, dst, acc, (int)nTiles);
    if (rem > 0)
      brain_edges_tail<<<(int)((rem + tpb - 1) / tpb), tpb, 0, stream>>>(
          v, wgt, src, dst, acc, tail0, E);
    add_bias<<<(int)((NB + tpb - 1) / tpb), tpb, 0, stream>>>(inon, bias, acc, (int)NB);
    update_state<<<(int)((N + tpb - 1) / tpb), tpb, 0, stream>>>(v, acc, msk, (int)N);
  }
  gather_out<<<(int)((NO + tpb - 1) / tpb), tpb, 0, stream>>>(v, iout, out, (int)NO);
}